// PAdicNeuralNetwork_8924942041531
// MI455X (gfx1250) — compile-verified
//
#include <hip/hip_runtime.h>
#include <math.h>

// ---------------------------------------------------------------------------
// MI455X / gfx1250 implementation of the p-adic network forward pass.
// Dominant cost: the 537MB `ultra` tensor store (~23us at 23.3 TB/s HBM).
// All GEMMs use V_WMMA_F32_16X16X4_F32 (f32, matches reference precision).
// ---------------------------------------------------------------------------

typedef float v2f __attribute__((ext_vector_type(2)));
typedef float v4f __attribute__((ext_vector_type(4)));
typedef float v8f __attribute__((ext_vector_type(8)));

constexpr int kBatch = 4;
constexpr int kSeq   = 512;
constexpr int kE     = 128;
constexpr int kHD    = 32;

__device__ __forceinline__ float padic_act(float x) {
  // sign(x) * 2^floor(log2(|x| + 1e-9))
  float mag = fabsf(x);
  float sgn = (x > 0.f) ? 1.f : ((x < 0.f) ? -1.f : 0.f);
  float lvl = floorf(__log2f(mag + 1e-9f));
  return exp2f(lvl) * sgn;
}

__device__ __forceinline__ float gelu_exact(float x) {
  return 0.5f * x * (1.f + erff(x * 0.70710678118654752f));
}

// ---------------------------------------------------------------------------
// Generic WMMA f32 GEMM: C[M,N] = act(A[M,K] @ B[K,N] + bias[N])
// One wave32 per 16x16 output tile. M,N multiples of 16; K multiple of 4.
// A-matrix lane layout (V_WMMA_F32_16X16X4_F32): lane m = L&15,
//   VGPR0 = K(2*(L>>4)), VGPR1 = K(2*(L>>4)+1). B symmetric with n = L&15.
// D layout: VGPR t holds (m = m0 + t + 8*(L>>4), n = n0 + (L&15)).
// ACT: 0 = bias only, 1 = p-adic, 2 = exact GELU.
// ---------------------------------------------------------------------------
template <int ACT>
__global__ __launch_bounds__(32) void gemm16(const float* __restrict__ A,
                                             const float* __restrict__ Bm,
                                             const float* __restrict__ bias,
                                             float* __restrict__ C,
                                             int M, int N, int K) {
  const int ntn = N >> 4;
  const int tm  = blockIdx.x / ntn;
  const int tn  = blockIdx.x - tm * ntn;
  const int lane = threadIdx.x;
  const int l15  = lane & 15;
  const int lh   = lane >> 4;
  const int m0 = tm << 4, n0 = tn << 4;

  const float* Arow = A + (size_t)(m0 + l15) * K;
  const float* Bcol = Bm + n0 + l15;

  v8f c = {0.f, 0.f, 0.f, 0.f, 0.f, 0.f, 0.f, 0.f};
  for (int k = 0; k < K; k += 4) {
    v2f a, b;
    a.x = Arow[k + 2 * lh];
    a.y = Arow[k + 2 * lh + 1];
    b.x = Bcol[(size_t)(k + 2 * lh) * N];
    b.y = Bcol[(size_t)(k + 2 * lh + 1) * N];
    c = __builtin_amdgcn_wmma_f32_16x16x4_f32(false, a, false, b, (short)0, c,
                                              false, false);
  }

  const float bv = bias[n0 + l15];
#pragma unroll
  for (int t = 0; t < 8; ++t) {
    float y = c[t] + bv;
    if (ACT == 1) y = padic_act(y);
    if (ACT == 2) y = gelu_exact(y);
    C[(size_t)(m0 + t + 8 * lh) * N + n0 + l15] = y;
  }
}

// ---------------------------------------------------------------------------
// Embedding gather + per-channel scale: h0[row, e] = embed[x[row], e]*scale[e]
// ---------------------------------------------------------------------------
__global__ __launch_bounds__(256) void embed_kernel(
    const int* __restrict__ x, const float* __restrict__ emb,
    const float* __restrict__ scale, float* __restrict__ h0) {
  int idx = blockIdx.x * 256 + threadIdx.x;  // (B*S*E) threads
  int row = idx >> 7, e = idx & 127;
  h0[idx] = emb[(size_t)x[row] * kE + e] * scale[e];
}

// ---------------------------------------------------------------------------
// p-adic attention for one (batch, head, 16-row i-tile).
//   dists[i,j] = max_d |q[i,d] - k[j,d]|  (VALU, staged in LDS)
//   softmax(-dists) in place, then attended = attn @ V via WMMA (K=512).
// 128 threads = 4 waves; waves 0/1 run the two 16-col WMMA n-tiles (HD=32).
// ---------------------------------------------------------------------------
__global__ __launch_bounds__(128) void padic_attn(const float* __restrict__ q,
                                                  const float* __restrict__ k,
                                                  const float* __restrict__ vv,
                                                  float* __restrict__ att) {
  __shared__ float sattn[16][kSeq];   // 32 KB
  __shared__ float sq[16][kHD];       // 2 KB
  const int bid   = blockIdx.x;       // B*NH*(S/16) = 512
  const int itile = bid & 31;
  const int head  = (bid >> 5) & 3;
  const int batch = bid >> 7;
  const int tid   = threadIdx.x;
  const int base  = batch * kSeq;

  for (int idx = tid; idx < 16 * kHD; idx += 128) {
    int i = idx >> 5, d = idx & 31;
    sq[i][d] = q[(size_t)(base + itile * 16 + i) * kE + head * kHD + d];
  }
  __syncthreads();

  for (int idx = tid; idx < 16 * kSeq; idx += 128) {
    int i = idx >> 9, j = idx & 511;
    const float* kr = k + (size_t)(base + j) * kE + head * kHD;
    float m = 0.f;
#pragma unroll 8
    for (int d = 0; d < kHD; ++d) m = fmaxf(m, fabsf(sq[i][d] - kr[d]));
    sattn[i][j] = m;
  }
  __syncthreads();

  const int wave = tid >> 5, lane = tid & 31;
  for (int i = wave; i < 16; i += 4) {
    float mx = -3.4e38f;
    for (int j = lane; j < kSeq; j += 32) mx = fmaxf(mx, -sattn[i][j]);
#pragma unroll
    for (int off = 16; off > 0; off >>= 1)
      mx = fmaxf(mx, __shfl_xor(mx, off, 32));
    float sum = 0.f;
    for (int j = lane; j < kSeq; j += 32) {
      float p = __expf(-sattn[i][j] - mx);
      sattn[i][j] = p;
      sum += p;
    }
#pragma unroll
    for (int off = 16; off > 0; off >>= 1) sum += __shfl_xor(sum, off, 32);
    float inv = 1.f / sum;
    for (int j = lane; j < kSeq; j += 32) sattn[i][j] *= inv;
  }
  __syncthreads();

  if (wave < 2) {  // whole-wave uniform branch: EXEC all-ones for WMMA
    const int n0 = wave * 16;
    const int l15 = lane & 15, lh = lane >> 4;
    v8f c = {0.f, 0.f, 0.f, 0.f, 0.f, 0.f, 0.f, 0.f};
    for (int kk = 0; kk < kSeq; kk += 4) {
      v2f a, b;
      a.x = sattn[l15][kk + 2 * lh];
      a.y = sattn[l15][kk + 2 * lh + 1];
      const float* vr =
          vv + (size_t)(base + kk + 2 * lh) * kE + head * kHD + n0 + l15;
      b.x = vr[0];
      b.y = vr[kE];
      c = __builtin_amdgcn_wmma_f32_16x16x4_f32(false, a, false, b, (short)0, c,
                                                false, false);
    }
#pragma unroll
    for (int t = 0; t < 8; ++t) {
      int i = itile * 16 + t + 8 * lh;
      att[(size_t)(base + i) * kE + head * kHD + n0 + l15] = c[t];
    }
  }
}

// ---------------------------------------------------------------------------
// out = LayerNorm(x + r) * g + b ; one wave per 128-wide row, 8 rows/block.
// ---------------------------------------------------------------------------
__global__ __launch_bounds__(256) void residual_ln(
    const float* __restrict__ x, const float* __restrict__ r,
    const float* __restrict__ g, const float* __restrict__ bb,
    float* __restrict__ out) {
  int row = blockIdx.x * 8 + (threadIdx.x >> 5);
  int lane = threadIdx.x & 31;
  const float* xr = x + (size_t)row * kE;
  const float* rr = r + (size_t)row * kE;
  float v[4];
  float s = 0.f;
#pragma unroll
  for (int t = 0; t < 4; ++t) {
    v[t] = xr[lane + 32 * t] + rr[lane + 32 * t];
    s += v[t];
  }
#pragma unroll
  for (int off = 16; off > 0; off >>= 1) s += __shfl_xor(s, off, 32);
  float mu = s * (1.f / 128.f);
  float var = 0.f;
#pragma unroll
  for (int t = 0; t < 4; ++t) {
    float d = v[t] - mu;
    var += d * d;
  }
#pragma unroll
  for (int off = 16; off > 0; off >>= 1) var += __shfl_xor(var, off, 32);
  var *= (1.f / 128.f);
  float inv = rsqrtf(var + 1e-5f);
  float* orow = out + (size_t)row * kE;
#pragma unroll
  for (int t = 0; t < 4; ++t) {
    int e = lane + 32 * t;
    orow[e] = (v[t] - mu) * inv * g[e] + bb[e];
  }
}

// gr[b,e] = mean over 32 rows of cur (B,32,E)
__global__ __launch_bounds__(256) void mean_kernel(const float* __restrict__ c,
                                                   float* __restrict__ gr) {
  int idx = blockIdx.x * 256 + threadIdx.x;  // 512
  int b = idx >> 7, e = idx & 127;
  float s = 0.f;
#pragma unroll
  for (int r = 0; r < 32; ++r) s += c[(size_t)(b * 32 + r) * kE + e];
  gr[idx] = s * (1.f / 32.f);
}

// output[b,n] = gr[b,:] @ out_w[:,n] + out_b[n]  (4x128, too small for WMMA)
__global__ __launch_bounds__(256) void outproj_kernel(
    const float* __restrict__ gr, const float* __restrict__ w,
    const float* __restrict__ bb, float* __restrict__ o) {
  int idx = blockIdx.x * 256 + threadIdx.x;  // 512
  int b = idx >> 7, n = idx & 127;
  float acc = bb[n];
  for (int k2 = 0; k2 < kE; ++k2) acc += gr[b * kE + k2] * w[k2 * kE + n];
  o[idx] = acc;
}

// ---------------------------------------------------------------------------
// ultra[b,i,j,e] = |h[b,i,e] - h[b,j,e]|  -- 537 MB output stream.
// 128-bit loads (h stays L2-resident, 1 MB) + non-temporal 128-bit stores.
// Flat-index trick: output offset == global thread id * 4 floats.
// ---------------------------------------------------------------------------
__global__ __launch_bounds__(256) void ultra_kernel(const float* __restrict__ h,
                                                    float* __restrict__ ultra) {
  unsigned idx = blockIdx.x * 256u + threadIdx.x;  // 2^25 threads
  unsigned e4 = idx & 31u;
  unsigned j  = (idx >> 5) & 511u;
  unsigned i  = (idx >> 14) & 511u;
  unsigned b  = idx >> 23;
  const v4f hi = *(const v4f*)(h + (size_t)(b * kSeq + i) * kE + e4 * 4);
  const v4f hj = *(const v4f*)(h + (size_t)(b * kSeq + j) * kE + e4 * 4);
  v4f d;
  d.x = fabsf(hi.x - hj.x);
  d.y = fabsf(hi.y - hj.y);
  d.z = fabsf(hi.z - hj.z);
  d.w = fabsf(hi.w - hj.w);
  __builtin_nontemporal_store(d, (v4f*)(ultra + (size_t)idx * 4));
}

// ---------------------------------------------------------------------------
extern "C" void kernel_launch(void* const* d_in, const int* in_sizes, int n_in,
                              void* d_out, int out_size, void* d_ws,
                              size_t ws_size, hipStream_t stream) {
  (void)in_sizes; (void)n_in; (void)out_size; (void)ws_size;

  const int*   x     = (const int*)d_in[0];
  const float* embed = (const float*)d_in[1];
  const float* scale = (const float*)d_in[2];
  const float* q_w = (const float*)d_in[3],  *q_b = (const float*)d_in[4];
  const float* k_w = (const float*)d_in[5],  *k_b = (const float*)d_in[6];
  const float* v_w = (const float*)d_in[7],  *v_b = (const float*)d_in[8];
  const float* o_w = (const float*)d_in[9],  *o_b = (const float*)d_in[10];
  const float* ff1_w = (const float*)d_in[11], *ff1_b = (const float*)d_in[12];
  const float* ff2_w = (const float*)d_in[13], *ff2_b = (const float*)d_in[14];
  const float* agg_w[4] = {(const float*)d_in[15], (const float*)d_in[16],
                           (const float*)d_in[17], (const float*)d_in[18]};
  const float* agg_b[4] = {(const float*)d_in[19], (const float*)d_in[20],
                           (const float*)d_in[21], (const float*)d_in[22]};
  const float* ln1_g = (const float*)d_in[23], *ln1_b = (const float*)d_in[24];
  const float* ln2_g = (const float*)d_in[25], *ln2_b = (const float*)d_in[26];
  const float* out_w = (const float*)d_in[27], *out_b = (const float*)d_in[28];

  float* out = (float*)d_out;
  float* ws  = (float*)d_ws;

  // workspace layout (floats); total ~13.6 MB
  float* h0   = ws;                // 2048*128
  float* qb   = ws + 262144;
  float* kb   = ws + 524288;
  float* vb   = ws + 786432;
  float* attb = ws + 1048576;
  float* atto = ws + 1310720;
  float* h1   = ws + 1572864;
  float* ff1o = ws + 1835008;      // 2048*512
  float* ff2o = ws + 2883584;
  float* trA  = ws + 3145728;      // <= 1024*128
  float* trB  = ws + 3276800;

  // d_out layout: output(512) | h(262144) | global_repr(512) | ultra(2^27)
  float* out_final = out;
  float* h_out     = out + 512;
  float* gr        = out + 262656;
  float* ultra     = out + 263168;

  embed_kernel<<<1024, 256, 0, stream>>>(x, embed, scale, h0);

  gemm16<0><<<1024, 32, 0, stream>>>(h0, q_w, q_b, qb, 2048, 128, 128);
  gemm16<0><<<1024, 32, 0, stream>>>(h0, k_w, k_b, kb, 2048, 128, 128);
  gemm16<0><<<1024, 32, 0, stream>>>(h0, v_w, v_b, vb, 2048, 128, 128);

  padic_attn<<<512, 128, 0, stream>>>(qb, kb, vb, attb);

  gemm16<0><<<1024, 32, 0, stream>>>(attb, o_w, o_b, atto, 2048, 128, 128);
  residual_ln<<<256, 256, 0, stream>>>(h0, atto, ln1_g, ln1_b, h1);

  gemm16<1><<<4096, 32, 0, stream>>>(h1, ff1_w, ff1_b, ff1o, 2048, 512, 128);
  gemm16<0><<<1024, 32, 0, stream>>>(ff1o, ff2_w, ff2_b, ff2o, 2048, 128, 512);
  residual_ln<<<256, 256, 0, stream>>>(h1, ff2o, ln2_g, ln2_b, h_out);

  // Tree aggregation: concat(even,odd) is a contiguous (M, 2E) view.
  gemm16<2><<<512, 32, 0, stream>>>(h_out, agg_w[0], agg_b[0], trA, 1024, 128, 256);
  gemm16<2><<<256, 32, 0, stream>>>(trA, agg_w[1], agg_b[1], trB, 512, 128, 256);
  gemm16<2><<<128, 32, 0, stream>>>(trB, agg_w[2], agg_b[2], trA, 256, 128, 256);
  gemm16<2><<<64, 32, 0, stream>>>(trA, agg_w[3], agg_b[3], trB, 128, 128, 256);

  mean_kernel<<<2, 256, 0, stream>>>(trB, gr);
  outproj_kernel<<<2, 256, 0, stream>>>(gr, out_w, out_b, out_final);

  ultra_kernel<<<131072, 256, 0, stream>>>(h_out, ultra);
}